// G2pEncoder_51505247813893
// MI455X (gfx1250) — compile-verified
//
#include <hip/hip_runtime.h>
#include <hip/hip_bf16.h>

typedef __attribute__((ext_vector_type(2))) float v2f;
typedef __attribute__((ext_vector_type(8))) float v8f;

#define E_DIM 1024
#define H_DIM 2048
#define G3H   6144          // 3*H
#define T_MAX 1025
#define DEC_E 1024

// workspace layout (bytes)
#define WS_CNT   0
#define WS_H     64
#define WS_RZNHH (WS_H + H_DIM * 4)          // 8256
#define WS_RZNIH (WS_RZNHH + G3H * 4)        // 32832 ; + 1025*6144*4 ~= 25.2 MB total

#define SCAN_BLOCKS 64
#define SCAN_TPB    256

// ---------------------------------------------------------------------------
// Kernel 0: per-call init (deterministic): zero barrier counter, zero h0,
// copy dec_emb[2] into d_out[0:1024].
// ---------------------------------------------------------------------------
__global__ void g2p_init_kernel(const float* __restrict__ dec_emb,
                                float* __restrict__ out_dec,
                                float* __restrict__ h_g,
                                unsigned* __restrict__ cnt) {
  int g = blockIdx.x * blockDim.x + threadIdx.x;
  if (g == 0) *cnt = 0u;
  if (g < H_DIM) h_g[g] = 0.0f;
  if (g < DEC_E) out_dec[g] = dec_emb[2 * E_DIM + g];
}

// ---------------------------------------------------------------------------
// Kernel 1: fused embedding-gather + WMMA f32 GEMM.
// rzn_ih[t, j] = sum_e enc_emb[x[t], e] * w_ih[j, e] + b_ih[j]
// One wave computes a 16(M=t) x 64(N=j) tile. V_WMMA_F32_16X16X4_F32 layout:
//   A 16x4 : lane<16 holds A[M=lane][K=0..1] in v0..v1, lane>=16 holds K=2..3
//   B 4x16 : lane<16 holds B[K=0..1][N=lane], lane>=16 holds K=2..3
//   C 16x16: vgpr g -> row g + 8*(lane/16), col = lane%16
// ---------------------------------------------------------------------------
__global__ void __launch_bounds__(32)
g2p_ih_gemm_wmma(const int* __restrict__ x,
                 const float* __restrict__ enc_emb,
                 const float* __restrict__ w_ih,
                 const float* __restrict__ b_ih,
                 float* __restrict__ rzn_ih) {
  const int lane = threadIdx.x;
  const int half = lane >> 4;          // 0/1 -> K pair 0..1 / 2..3
  const int lid  = lane & 15;          // M for A-load, N for B-load / C-store
  const int t0   = blockIdx.y * 16;
  const int n0   = blockIdx.x * 64;

  // A: gathered embedding row for this lane's M (clamp tail tile)
  int t   = t0 + lid;
  int tok = x[t < T_MAX ? t : (T_MAX - 1)];
  const float* arow = enc_emb + (size_t)tok * E_DIM + 2 * half;

  // B: w_ih rows (row-major [6144 x 1024]); B[k][n] = w_ih[n0+n][kbase+k]
  const float* brow = w_ih + (size_t)(n0 + lid) * E_DIM + 2 * half;

  v8f c0 = {}, c1 = {}, c2 = {}, c3 = {};
  for (int k = 0; k < E_DIM; k += 4) {
    v2f a  = *(const v2f*)(arow + k);
    v2f b0 = *(const v2f*)(brow + k);
    v2f b1 = *(const v2f*)(brow + 16 * E_DIM + k);
    v2f b2 = *(const v2f*)(brow + 32 * E_DIM + k);
    v2f b3 = *(const v2f*)(brow + 48 * E_DIM + k);
    c0 = __builtin_amdgcn_wmma_f32_16x16x4_f32(false, a, false, b0, (short)0, c0, false, false);
    c1 = __builtin_amdgcn_wmma_f32_16x16x4_f32(false, a, false, b1, (short)0, c1, false, false);
    c2 = __builtin_amdgcn_wmma_f32_16x16x4_f32(false, a, false, b2, (short)0, c2, false, false);
    c3 = __builtin_amdgcn_wmma_f32_16x16x4_f32(false, a, false, b3, (short)0, c3, false, false);
  }

  const float bias0 = b_ih[n0 + lid];
  const float bias1 = b_ih[n0 + 16 + lid];
  const float bias2 = b_ih[n0 + 32 + lid];
  const float bias3 = b_ih[n0 + 48 + lid];

#pragma unroll
  for (int g = 0; g < 8; ++g) {
    int row = t0 + g + 8 * half;
    if (row < T_MAX) {
      float* o = rzn_ih + (size_t)row * G3H + n0 + lid;
      o[0]  = c0[g] + bias0;
      o[16] = c1[g] + bias1;
      o[32] = c2[g] + bias2;
      o[48] = c3[g] + bias3;
    }
  }
}

// ---------------------------------------------------------------------------
// Device-wide barrier (atomic counter, monotonically increasing phase).
// ---------------------------------------------------------------------------
__device__ __forceinline__ void grid_barrier(unsigned* cnt, unsigned& ph, unsigned nblk) {
  __scoped_atomic_thread_fence(__ATOMIC_RELEASE, __MEMORY_SCOPE_DEVICE);
  __syncthreads();
  ++ph;
  if (threadIdx.x == 0) {
    __hip_atomic_fetch_add(cnt, 1u, __ATOMIC_RELAXED, __HIP_MEMORY_SCOPE_AGENT);
    while (__hip_atomic_load(cnt, __ATOMIC_RELAXED, __HIP_MEMORY_SCOPE_AGENT) < ph * nblk) {
      __builtin_amdgcn_s_sleep(1);
    }
  }
  __syncthreads();
  __scoped_atomic_thread_fence(__ATOMIC_ACQUIRE, __MEMORY_SCOPE_DEVICE);
}

// ---------------------------------------------------------------------------
// Kernel 2: persistent GRU scan. Per step: whole-device GEMV
// rzn_hh = w_hh @ h + b_hh (wave-per-row, h staged in LDS), barrier,
// gate update on threads 0..2047, barrier.
// ---------------------------------------------------------------------------
__global__ void __launch_bounds__(SCAN_TPB)
g2p_gru_scan(const int* __restrict__ len_words,
             const float* __restrict__ w_hh,
             const float* __restrict__ b_hh,
             const float* __restrict__ rzn_ih,
             float* __restrict__ h_g,
             float* __restrict__ rzn_hh,
             unsigned* __restrict__ cnt,
             float* __restrict__ out_h) {
  __shared__ float h_s[H_DIM];
  const int tid    = threadIdx.x;
  const int gid    = blockIdx.x * SCAN_TPB + tid;
  const int lane   = tid & 31;
  const int wave_g = gid >> 5;                          // 0..511
  const int nwaves = (SCAN_BLOCKS * SCAN_TPB) >> 5;     // 512

  int steps = *len_words + 1;
  if (steps > T_MAX) steps = T_MAX;
  unsigned ph = 0;

  for (int t = 0; t < steps; ++t) {
    // stage h into LDS (8 KB), coalesced
#pragma unroll
    for (int i = tid; i < H_DIM; i += SCAN_TPB) h_s[i] = h_g[i];
    __syncthreads();

    // GEMV: each wave owns rows j = wave_g + 512*rep (12 rows per wave)
#pragma unroll
    for (int rep = 0; rep < G3H / (SCAN_BLOCKS * SCAN_TPB / 32); ++rep) {
      const int j = wave_g + nwaves * rep;
      const float4* wrow = (const float4*)(w_hh + (size_t)j * H_DIM);
      const float4* hv4  = (const float4*)h_s;
      float acc = 0.0f;
#pragma unroll
      for (int i2 = 0; i2 < H_DIM / 128; ++i2) {        // 16 x b128 per lane
        float4 w4 = wrow[lane + 32 * i2];
        float4 h4 = hv4[lane + 32 * i2];
        acc = fmaf(w4.x, h4.x, acc);
        acc = fmaf(w4.y, h4.y, acc);
        acc = fmaf(w4.z, h4.z, acc);
        acc = fmaf(w4.w, h4.w, acc);
      }
#pragma unroll
      for (int m = 16; m >= 1; m >>= 1) acc += __shfl_xor(acc, m, 32);
      if (lane == 0) rzn_hh[j] = acc + b_hh[j];
    }

    grid_barrier(cnt, ph, SCAN_BLOCKS);

    // gate update (exactly matches reference formulas)
    if (gid < H_DIM) {
      const float* ih = rzn_ih + (size_t)t * G3H;
      float hr = rzn_hh[gid];
      float hz = rzn_hh[H_DIM + gid];
      float hn = rzn_hh[2 * H_DIM + gid];
      float r  = 1.0f / (1.0f + expf(-(ih[gid] + hr)));
      float z  = 1.0f / (1.0f + expf(-(ih[H_DIM + gid] + hz)));
      float n  = tanhf(ih[2 * H_DIM + gid] + r * hn);
      h_g[gid] = (1.0f - z) * n + z * h_g[gid];
    }

    grid_barrier(cnt, ph, SCAN_BLOCKS);
  }

  if (gid < H_DIM) out_h[gid] = h_g[gid];
}

// ---------------------------------------------------------------------------
extern "C" void kernel_launch(void* const* d_in, const int* in_sizes, int n_in,
                              void* d_out, int out_size, void* d_ws, size_t ws_size,
                              hipStream_t stream) {
  (void)in_sizes; (void)n_in; (void)out_size; (void)ws_size;
  const int*   x       = (const int*)  d_in[0];
  const int*   lenw    = (const int*)  d_in[1];
  const float* enc_emb = (const float*)d_in[2];
  const float* w_ih    = (const float*)d_in[3];
  const float* w_hh    = (const float*)d_in[4];
  const float* b_ih    = (const float*)d_in[5];
  const float* b_hh    = (const float*)d_in[6];
  const float* dec_emb = (const float*)d_in[7];
  float* out = (float*)d_out;

  char* ws = (char*)d_ws;
  unsigned* cnt  = (unsigned*)(ws + WS_CNT);
  float* h_g     = (float*)(ws + WS_H);
  float* rznhh   = (float*)(ws + WS_RZNHH);
  float* rznih   = (float*)(ws + WS_RZNIH);   // 1025*6144 f32 (~25.2 MB)

  // 0) deterministic per-call init
  g2p_init_kernel<<<8, 256, 0, stream>>>(dec_emb, out, h_g, cnt);

  // 1) hoisted input-side GEMM via native-f32 WMMA
  dim3 g1(G3H / 64, (T_MAX + 15) / 16);   // 96 x 65 waves
  g2p_ih_gemm_wmma<<<g1, 32, 0, stream>>>(x, enc_emb, w_ih, b_ih, rznih);

  // 2) persistent sequential scan (L2-resident w_hh, device-wide barriers)
  g2p_gru_scan<<<SCAN_BLOCKS, SCAN_TPB, 0, stream>>>(
      lenw, w_hh, b_hh, rznih, h_g, rznhh, cnt, out + DEC_E);
}